// MultiHeadHighLevelAllocator_27444841021902
// MI455X (gfx1250) — compile-verified
//
#include <hip/hip_runtime.h>

typedef __attribute__((ext_vector_type(2))) float v2f;
typedef __attribute__((ext_vector_type(8))) float v8f;

#define U_N     256
#define T_N     512
#define H_N     4
#define UAVD    64
#define TASKD   32
#define E_N     128
#define HID_N   256

// ---------------------------------------------------------------------------
// fp32 WMMA helper: accumulate a 16x16 tile of  X(16xK) @ W(NxK)^T  over K,
// A read from LDS (stride lda), B row pointer pre-offset by n*ldw.
// Layout per CDNA5 ISA 7.12.2 (16x16x4 f32).
// ---------------------------------------------------------------------------
__device__ __forceinline__ v8f wmma_k_loop(const float* __restrict__ ap,
                                           const float* __restrict__ bp,
                                           int K, v8f acc) {
  for (int kk = 0; kk < K; kk += 4) {
    v2f a; a.x = ap[kk]; a.y = ap[kk + 1];
    v2f b; b.x = bp[kk]; b.y = bp[kk + 1];
    acc = __builtin_amdgcn_wmma_f32_16x16x4_f32(false, a, false, b,
                                                (short)0, acc, false, false);
  }
  return acc;
}

// ---------------------------------------------------------------------------
// 3-layer MLP encoder:  relu(x@w0^T+b0) -> relu(@w1^T+b1) -> @w2^T+b2
// One block (4 waves) handles 16 rows; activations staged in LDS.
// ---------------------------------------------------------------------------
template <int INK>
__global__ __launch_bounds__(128) void encoder_kernel(
    const float* __restrict__ x,
    const float* __restrict__ w0, const float* __restrict__ b0,
    const float* __restrict__ w1, const float* __restrict__ b1,
    const float* __restrict__ w2, const float* __restrict__ b2,
    float* __restrict__ out) {
  __shared__ __align__(16) float bufA[16 * 132];
  __shared__ __align__(16) float bufB[16 * 132];
  const int tid  = threadIdx.x;
  const int lane = tid & 31;
  const int wave = tid >> 5;
  const int row0 = blockIdx.x * 16;

  // stage 16 x INK input tile
  for (int i = tid; i < 16 * INK; i += 128) {
    int r = i / INK, c = i - r * INK;
    bufA[r * 132 + c] = x[(size_t)(row0 + r) * INK + c];
  }
  __syncthreads();

  const int m    = lane & 15;            // row within tile (also col n)
  const int kb   = (lane >> 4) << 1;     // k sub-offset 0 or 2
  const int moff = (lane >> 4) * 8;      // D-row offset

  // ---- layer 1: bufA -> bufB, K = INK, relu ----
  #pragma unroll
  for (int t = 0; t < 2; ++t) {
    const int nb = (wave * 2 + t) * 16;
    v8f acc = {};
    acc = wmma_k_loop(bufA + m * 132 + kb,
                      w0 + (size_t)(nb + m) * INK + kb, INK, acc);
    const float bv = b0[nb + m];
    #pragma unroll
    for (int v = 0; v < 8; ++v) {
      float val = acc[v] + bv;
      bufB[(moff + v) * 132 + nb + m] = fmaxf(val, 0.f);
    }
  }
  __syncthreads();

  // ---- layer 2: bufB -> bufA, K = 128, relu ----
  #pragma unroll
  for (int t = 0; t < 2; ++t) {
    const int nb = (wave * 2 + t) * 16;
    v8f acc = {};
    acc = wmma_k_loop(bufB + m * 132 + kb,
                      w1 + (size_t)(nb + m) * E_N + kb, E_N, acc);
    const float bv = b1[nb + m];
    #pragma unroll
    for (int v = 0; v < 8; ++v) {
      float val = acc[v] + bv;
      bufA[(moff + v) * 132 + nb + m] = fmaxf(val, 0.f);
    }
  }
  __syncthreads();

  // ---- layer 3: bufA -> global, K = 128, no relu ----
  #pragma unroll
  for (int t = 0; t < 2; ++t) {
    const int nb = (wave * 2 + t) * 16;
    v8f acc = {};
    acc = wmma_k_loop(bufA + m * 132 + kb,
                      w2 + (size_t)(nb + m) * E_N + kb, E_N, acc);
    const float bv = b2[nb + m];
    #pragma unroll
    for (int v = 0; v < 8; ++v)
      out[(size_t)(row0 + moff + v) * E_N + nb + m] = acc[v] + bv;
  }
}

// ---------------------------------------------------------------------------
// pre_t[t][k] = task_embed[t] . Wt[k] + fb0[k],  Wt = fw0[:, E:]
// M=512, N=256, K=128. One wave per 16x16 tile, 8 waves/block.
// ---------------------------------------------------------------------------
__global__ __launch_bounds__(256) void pret_kernel(
    const float* __restrict__ taskE, const float* __restrict__ fw0,
    const float* __restrict__ fb0, float* __restrict__ preT) {
  const int lane  = threadIdx.x & 31;
  const int wave  = threadIdx.x >> 5;
  const int tile  = blockIdx.x * 8 + wave;
  const int ntile = tile & 15;   // 256/16
  const int ttile = tile >> 4;   // 512/16
  const int m    = lane & 15;
  const int kb   = (lane >> 4) << 1;
  const int moff = (lane >> 4) * 8;
  const int nb   = ntile * 16;

  v8f acc = {};
  acc = wmma_k_loop(taskE + (size_t)(ttile * 16 + m) * E_N + kb,
                    fw0 + (size_t)(nb + m) * (2 * E_N) + E_N + kb, E_N, acc);
  const float bv = fb0[nb + m];
  #pragma unroll
  for (int v = 0; v < 8; ++v)
    preT[(size_t)(ttile * 16 + moff + v) * HID_N + nb + m] = acc[v] + bv;
}

// ---------------------------------------------------------------------------
// pre_u[h][u][k] = (uav_embed[u] + head_queries[h]) . Wu[k],  Wu = fw0[:, :E]
// Virtual M = H*U = 1024, N=256, K=128.
// ---------------------------------------------------------------------------
__global__ __launch_bounds__(256) void preu_kernel(
    const float* __restrict__ uavE, const float* __restrict__ hq,
    const float* __restrict__ fw0, float* __restrict__ preU) {
  const int lane  = threadIdx.x & 31;
  const int wave  = threadIdx.x >> 5;
  const int tile  = blockIdx.x * 8 + wave;
  const int ntile = tile & 15;        // N tiles
  const int mtile = tile >> 4;        // 0..63
  const int h     = mtile >> 4;       // 16 u-tiles per head
  const int utile = mtile & 15;
  const int m    = lane & 15;
  const int kb   = (lane >> 4) << 1;
  const int moff = (lane >> 4) * 8;
  const int nb   = ntile * 16;

  const float* up = uavE + (size_t)(utile * 16 + m) * E_N + kb;
  const float* qp = hq + (size_t)h * E_N + kb;
  const float* bp = fw0 + (size_t)(nb + m) * (2 * E_N) + kb;

  v8f acc = {};
  for (int kk = 0; kk < E_N; kk += 4) {
    v2f a; a.x = up[kk] + qp[kk]; a.y = up[kk + 1] + qp[kk + 1];
    v2f b; b.x = bp[kk]; b.y = bp[kk + 1];
    acc = __builtin_amdgcn_wmma_f32_16x16x4_f32(false, a, false, b,
                                                (short)0, acc, false, false);
  }
  #pragma unroll
  for (int v = 0; v < 8; ++v)
    preU[((size_t)h * U_N + utile * 16 + moff + v) * HID_N + nb + m] = acc[v];
}

// ---------------------------------------------------------------------------
// out[h,u,t] = fb1 + sum_k relu(pre_u[h,u,k] + pre_t[t,k]) * fw1[k]
// ReLU breaks linearity -> VALU reduction; LDS tiles reused 16x each.
// Stride 260 floats => conflict-free float4 reads across 16 lanes.
// ---------------------------------------------------------------------------
__global__ __launch_bounds__(256) void logits_kernel(
    const float* __restrict__ preU, const float* __restrict__ preT,
    const float* __restrict__ fw1, const float* __restrict__ fb1,
    float* __restrict__ out) {
  __shared__ __align__(16) float puL[16 * 260];
  __shared__ __align__(16) float ptL[16 * 260];
  __shared__ __align__(16) float wL[HID_N];

  const int ttile = blockIdx.x & 31;
  const int utile = (blockIdx.x >> 5) & 15;
  const int h     = blockIdx.x >> 9;
  const int tid   = threadIdx.x;

  for (int i = tid; i < 16 * (HID_N / 4); i += 256) {
    int r = i >> 6, c4 = (i & 63) << 2;
    float4 a = *(const float4*)&preU[((size_t)h * U_N + utile * 16 + r) * HID_N + c4];
    *(float4*)&puL[r * 260 + c4] = a;
    float4 b = *(const float4*)&preT[((size_t)ttile * 16 + r) * HID_N + c4];
    *(float4*)&ptL[r * 260 + c4] = b;
  }
  if (tid < HID_N / 4)
    *(float4*)&wL[tid * 4] = *(const float4*)&fw1[tid * 4];
  __syncthreads();

  const int tt = tid & 15;
  const int tu = tid >> 4;
  const float* pu = &puL[tu * 260];
  const float* pt = &ptL[tt * 260];

  float acc = 0.f;
  #pragma unroll 4
  for (int k = 0; k < HID_N; k += 4) {
    float4 a = *(const float4*)&pu[k];
    float4 b = *(const float4*)&pt[k];
    float4 w = *(const float4*)&wL[k];
    acc = fmaf(fmaxf(a.x + b.x, 0.f), w.x, acc);
    acc = fmaf(fmaxf(a.y + b.y, 0.f), w.y, acc);
    acc = fmaf(fmaxf(a.z + b.z, 0.f), w.z, acc);
    acc = fmaf(fmaxf(a.w + b.w, 0.f), w.w, acc);
  }
  out[((size_t)h * U_N + utile * 16 + tu) * T_N + ttile * 16 + tt] = acc + fb1[0];
}

// ---------------------------------------------------------------------------
extern "C" void kernel_launch(void* const* d_in, const int* in_sizes, int n_in,
                              void* d_out, int out_size, void* d_ws, size_t ws_size,
                              hipStream_t stream) {
  (void)in_sizes; (void)n_in; (void)out_size; (void)ws_size;
  const float* uav_feat = (const float*)d_in[0];
  const float* task_feat = (const float*)d_in[1];
  const float* uw0 = (const float*)d_in[2];
  const float* ub0 = (const float*)d_in[3];
  const float* uw1 = (const float*)d_in[4];
  const float* ub1 = (const float*)d_in[5];
  const float* uw2 = (const float*)d_in[6];
  const float* ub2 = (const float*)d_in[7];
  const float* tw0 = (const float*)d_in[8];
  const float* tb0 = (const float*)d_in[9];
  const float* tw1 = (const float*)d_in[10];
  const float* tb1 = (const float*)d_in[11];
  const float* tw2 = (const float*)d_in[12];
  const float* tb2 = (const float*)d_in[13];
  const float* head_queries = (const float*)d_in[14];
  const float* fw0 = (const float*)d_in[15];
  const float* fb0 = (const float*)d_in[16];
  const float* fw1 = (const float*)d_in[17];
  const float* fb1 = (const float*)d_in[18];
  float* out = (float*)d_out;

  float* ws = (float*)d_ws;
  float* uavE  = ws;                         // 256*128  = 32768
  float* taskE = ws + 32768;                 // 512*128  = 65536
  float* preT  = ws + 32768 + 65536;         // 512*256  = 131072
  float* preU  = ws + 32768 + 65536 + 131072;// 4*256*256 = 262144

  encoder_kernel<UAVD><<<U_N / 16, 128, 0, stream>>>(
      uav_feat, uw0, ub0, uw1, ub1, uw2, ub2, uavE);
  encoder_kernel<TASKD><<<T_N / 16, 128, 0, stream>>>(
      task_feat, tw0, tb0, tw1, tb1, tw2, tb2, taskE);

  // pre_t: 512 tiles of 16x16, 8 waves/block -> 64 blocks
  pret_kernel<<<64, 256, 0, stream>>>(taskE, fw0, fb0, preT);
  // pre_u: 1024 tiles -> 128 blocks
  preu_kernel<<<128, 256, 0, stream>>>(uavE, head_queries, fw0, preU);
  // final: H * (U/16) * (T/16) = 2048 blocks
  logits_kernel<<<H_N * (U_N / 16) * (T_N / 16), 256, 0, stream>>>(
      preU, preT, fw1, fb1, out);
}